// FloodFillOperation_42580305773207
// MI455X (gfx1250) — compile-verified
//
#include <hip/hip_runtime.h>
#include <stdint.h>

#define BATCH 256
#define NCH   10
#define HH    256
#define WW    256
#define WPR   8            // 256 cols / 32 bits per word
#define CHUNK_ROWS 16      // rows staged per async chunk (16 KB)

// Low 32 bits of the flat (generic) address of an LDS object are the LDS byte
// offset (aperture layout: {SHARED_BASE, lds_offset}).
__device__ __forceinline__ uint32_t lds_off_u32(const void* p) {
  return (uint32_t)(uintptr_t)p;
}

// CDNA5 async copy global -> LDS (VGLOBAL GLOBAL_LOAD_ASYNC_TO_LDS_B128, ASYNCcnt)
__device__ __forceinline__ void async_g2l_b128(uint32_t lds_byte_off, const void* gaddr) {
  asm volatile("global_load_async_to_lds_b128 %0, %1, off"
               :: "v"(lds_byte_off), "v"(gaddr)
               : "memory");
}

__device__ __forceinline__ void wait_async0() {
  asm volatile("s_wait_asynccnt 0x0" ::: "memory");
}

// ---------------- Kernel 1: per-batch seed detect + LDS flood fill ----------
extern "C" __global__ __launch_bounds__(256)
void flood_fill_kernel(const float* __restrict__ grid,
                       const int* __restrict__ seed_y_p,
                       const int* __restrict__ seed_x_p,
                       uint32_t* __restrict__ maskBits,
                       int* __restrict__ seedColorOut)
{
  __shared__ __align__(16) float stage[CHUNK_ROWS * WW];  // 16 KB staging
  __shared__ uint32_t colorBits[HH * WPR];                // 8 KB: color mask bits
  __shared__ uint32_t cur[HH * WPR];                      // 8 KB: fill state bits
  __shared__ int      s_min;
  __shared__ uint32_t s_changed;

  const int tid  = threadIdx.x;
  const int lane = tid & 31;      // wave32
  const int wv   = tid >> 5;      // 8 waves per block
  const int b    = blockIdx.x;
  const int sy   = seed_y_p[0];
  const int sx   = seed_x_p[0];

  if (tid == 0) { s_min = 0x7fffffff; s_changed = 0u; }
  __syncthreads();
  if (tid < NCH) {
    float v = grid[(((size_t)b * NCH + tid) * HH + sy) * WW + sx];
    if (v > 0.5f) atomicMin(&s_min, tid);
  }
  __syncthreads();
  const int  seedc = (s_min == 0x7fffffff) ? 0 : s_min;   // argmax of bool = first true
  const bool has   = (s_min != 0x7fffffff);

  // ---- stream seed-color channel through LDS (async), ballot-compress to bits
  const float*   chan       = grid + ((size_t)b * NCH + seedc) * (size_t)(HH * WW);
  const uint32_t stage_base = lds_off_u32(stage);
  for (int ck = 0; ck < HH / CHUNK_ROWS; ++ck) {
    const float* src = chan + (size_t)ck * CHUNK_ROWS * WW;
    #pragma unroll
    for (int j = 0; j < 4; ++j) {                 // 256 thr * 4 * 16B = 16 KB
      uint32_t f = (uint32_t)(j * 256 + tid) * 4u;  // float index (16B aligned)
      async_g2l_b128(stage_base + f * 4u, src + f);
    }
    wait_async0();
    __syncthreads();
    #pragma unroll
    for (int q = 0; q < 16; ++q) {                // 128 words per chunk
      int k = wv * 16 + q;
      uint32_t bits = (uint32_t)__ballot(stage[k * 32 + lane] > 0.5f);
      if (lane == 0) colorBits[ck * 128 + k] = bits;
    }
    __syncthreads();                              // stage reusable next chunk
  }

  // ---- init fill state: only the seed bit (if any channel present)
  for (int w = tid; w < HH * WPR; w += 256) cur[w] = 0u;
  __syncthreads();
  if (tid == 0 && has) cur[sy * WPR + (sx >> 5)] = 1u << (sx & 31);
  __syncthreads();

  // ---- iterate to fixed point: vertical merge + horizontal run-closure, & color
  const int r = tid;                              // one row per thread
  uint32_t nw[WPR];
  #pragma unroll
  for (int w = 0; w < WPR; ++w) nw[w] = 0u;

  for (int iter = 0; iter < HH * WW; ++iter) {
    uint32_t old[WPR], col[WPR];
    #pragma unroll
    for (int w = 0; w < WPR; ++w) {
      old[w]      = cur[r * WPR + w];
      uint32_t up = (r > 0)      ? cur[(r - 1) * WPR + w] : 0u;
      uint32_t dn = (r < HH - 1) ? cur[(r + 1) * WPR + w] : 0u;
      col[w]      = colorBits[r * WPR + w];
      nw[w]       = (old[w] | up | dn) & col[w];
    }
    // horizontal closure within the row, register-resident
    bool rowchg = true;
    while (rowchg) {
      rowchg = false;
      uint32_t t[WPR];
      #pragma unroll
      for (int w = 0; w < WPR; ++w) {
        uint32_t li = (w > 0)       ? (nw[w - 1] >> 31) : 0u;
        uint32_t ri = (w < WPR - 1) ? (nw[w + 1] << 31) : 0u;
        t[w] = (nw[w] | (nw[w] << 1) | li | (nw[w] >> 1) | ri) & col[w];
      }
      #pragma unroll
      for (int w = 0; w < WPR; ++w) { rowchg |= (t[w] != nw[w]); nw[w] = t[w]; }
    }
    uint32_t diff = 0u;
    #pragma unroll
    for (int w = 0; w < WPR; ++w) diff |= (nw[w] ^ old[w]);

    __syncthreads();                              // all reads done before writes
    #pragma unroll
    for (int w = 0; w < WPR; ++w) cur[r * WPR + w] = nw[w];
    if (diff) atomicOr(&s_changed, 1u);
    __syncthreads();
    const bool again = (s_changed != 0u);
    __syncthreads();                              // everyone sampled the flag
    if (tid == 0) s_changed = 0u;
    if (!again) break;
  }

  // ---- emit mask bits (8 KB/batch) + seed color
  #pragma unroll
  for (int w = 0; w < WPR; ++w)
    maskBits[(size_t)b * (HH * WPR) + r * WPR + w] = nw[w];
  if (tid == 0) seedColorOut[b] = seedc;
}

// ---------------- Kernel 2: bandwidth-bound masked rewrite ------------------
extern "C" __global__ __launch_bounds__(256)
void rewrite_kernel(const float4* __restrict__ grid4,
                    const uint32_t* __restrict__ maskBits,
                    const int* __restrict__ seedColor,
                    const int* __restrict__ target_p,
                    float4* __restrict__ out4)
{
  const size_t N4 = (size_t)BATCH * NCH * HH * WW / 4;
  size_t i4 = (size_t)blockIdx.x * blockDim.x + threadIdx.x;
  if (i4 >= N4) return;
  __builtin_prefetch(grid4 + i4, 0, 0);           // global_prefetch_b8

  const int tgt = target_p[0];
  size_t e  = i4 * 4;
  int    p  = (int)(e & (size_t)(HH * WW - 1));   // spatial index (H*W = 65536)
  size_t bc = e >> 16;
  int    c  = (int)(bc % NCH);
  int    b  = (int)(bc / NCH);

  uint32_t wbits = maskBits[(size_t)b * (HH * WPR) + (p >> 5)];
  int      sc    = seedColor[b];
  float4   g     = grid4[i4];
  int      sh    = p & 31;                        // p % 4 == 0, so sh <= 28

  float r0 = g.x, r1 = g.y, r2 = g.z, r3 = g.w;
  if ((wbits >> (sh + 0)) & 1u) { if (c == sc) r0 = 0.f; if (c == tgt) r0 = 1.f; }
  if ((wbits >> (sh + 1)) & 1u) { if (c == sc) r1 = 0.f; if (c == tgt) r1 = 1.f; }
  if ((wbits >> (sh + 2)) & 1u) { if (c == sc) r2 = 0.f; if (c == tgt) r2 = 1.f; }
  if ((wbits >> (sh + 3)) & 1u) { if (c == sc) r3 = 0.f; if (c == tgt) r3 = 1.f; }
  out4[i4] = make_float4(r0, r1, r2, r3);
}

// ---------------- launcher --------------------------------------------------
extern "C" void kernel_launch(void* const* d_in, const int* in_sizes, int n_in,
                              void* d_out, int out_size, void* d_ws, size_t ws_size,
                              hipStream_t stream) {
  const float* grid   = (const float*)d_in[0];
  const int*   seed_y = (const int*)d_in[1];
  const int*   seed_x = (const int*)d_in[2];
  const int*   target = (const int*)d_in[3];
  float*       out    = (float*)d_out;

  int*      seedColor = (int*)d_ws;                        // 1 KB
  uint32_t* maskBits  = (uint32_t*)((char*)d_ws + 1024);   // 2 MB of mask bits

  flood_fill_kernel<<<BATCH, 256, 0, stream>>>(grid, seed_y, seed_x,
                                               maskBits, seedColor);

  const size_t N4 = (size_t)BATCH * NCH * HH * WW / 4;     // 41,943,040
  rewrite_kernel<<<(unsigned)(N4 / 256), 256, 0, stream>>>(
      (const float4*)grid, maskBits, seedColor, target, (float4*)out);
}